// NonMaximumSuppression_43276090475236
// MI455X (gfx1250) — compile-verified
//
#include <hip/hip_runtime.h>
#include <stdint.h>

// ---- problem constants (match reference) ----
#define NA     300000     // anchors
#define NC     80         // classes
#define DD     6          // detection feature dim
#define TK     8000       // top-k
#define TKP    8192       // padded top-k (power of two for bitonic sort)
#define MB     300        // MAX_BOXES
#define NBINS  4096       // radix-select bins (top 12 bits of flipped float)
#define NMS1_THR   0.4f
#define SCORE_THR  0.05f
#define NMS2_THR   0.05f
#define NEGF      (-1e30f)
#define PADF      (-1e38f)

typedef __attribute__((ext_vector_type(2))) float v2f;
typedef __attribute__((ext_vector_type(8))) float v8f;

// ---- workspace layout (bytes) ----
static constexpr size_t OFF_SCORES = 0;                       // NA*4 = 1200000
static constexpr size_t OFF_HIST   = 1200128;                 // 4096*4
static constexpr size_t OFF_META   = OFF_HIST + 16384;        // 16 u32
static constexpr size_t OFF_TKS    = OFF_META + 64;           // TKP*4
static constexpr size_t OFF_TKI    = OFF_TKS + 32768;         // TKP*4
static constexpr size_t OFF_TKB    = OFF_TKI + 32768;         // TKP*16 (16B aligned)
static constexpr size_t OFF_TKD    = OFF_TKB + 131072;        // TK*6*4
static constexpr size_t OFF_S1     = OFF_TKD + 192000;        // MB*4
static constexpr size_t OFF_B1     = OFF_S1 + 1280;           // MB*16 (16B aligned)
static constexpr size_t OFF_D1     = OFF_B1 + 4864;           // MB*6*4
static constexpr size_t OFF_IOU2   = OFF_D1 + 7232;           // MB*MB*4 = 360000

__device__ __forceinline__ float boxArea(float4 b) { return (b.z - b.x) * (b.w - b.y); }

// ---------------------------------------------------------------------------
// K0: zero histogram + meta counters, pad top-k arrays.
// ---------------------------------------------------------------------------
__global__ void init_ws_kernel(unsigned* hist, unsigned* meta, float* tks, int* tki) {
    int i = blockIdx.x * blockDim.x + threadIdx.x;
    if (i < NBINS) hist[i] = 0u;
    if (i < 16)    meta[i] = 0u;
    if (i < TKP) { tks[i] = PADF; tki[i] = 0; }
}

// ---------------------------------------------------------------------------
// K1: scores[a] = max_c classification[a,c]. 4 threads cooperate per anchor,
// each reading 5 float4s (64B contiguous per 4-lane group). Bandwidth stage.
// ---------------------------------------------------------------------------
__global__ void score_max_kernel(const float* __restrict__ cls, float* __restrict__ scores) {
    int gt = blockIdx.x * blockDim.x + threadIdx.x;
    int a = gt >> 2, sub = gt & 3;
    if (a >= NA) return;
    const float4* row = (const float4*)(cls + (size_t)a * NC); // 20 float4 per anchor
    __builtin_prefetch(row + 20 * 64, 0, 1);                   // global_prefetch_b8
    float m = PADF;
#pragma unroll
    for (int k = 0; k < 5; ++k) {
        float4 v = row[sub + 4 * k];
        m = fmaxf(m, fmaxf(fmaxf(v.x, v.y), fmaxf(v.z, v.w)));
    }
    m = fmaxf(m, __shfl_xor(m, 1, 4));
    m = fmaxf(m, __shfl_xor(m, 2, 4));
    if (sub == 0) scores[a] = m;
}

__device__ __forceinline__ unsigned orderKey(float s) {
    unsigned u = __float_as_uint(s);
    return (u & 0x80000000u) ? ~u : (u ^ 0x80000000u); // monotone float ordering
}

// ---------------------------------------------------------------------------
// K2: 12-bit histogram of order keys (radix-select for top-k threshold).
// ---------------------------------------------------------------------------
__global__ void hist_kernel(const float* __restrict__ scores, unsigned* __restrict__ hist) {
    int a = blockIdx.x * blockDim.x + threadIdx.x;
    if (a >= NA) return;
    atomicAdd(&hist[orderKey(scores[a]) >> 20], 1u);
}

// ---------------------------------------------------------------------------
// K3: serial suffix scan over 4096 bins (tiny) -> threshold bin.
// meta[0]=bin  meta[1]=count strictly above  meta[4]=needed from boundary bin
// ---------------------------------------------------------------------------
__global__ void threshold_kernel(const unsigned* __restrict__ hist, unsigned* __restrict__ meta) {
    if (threadIdx.x != 0 || blockIdx.x != 0) return;
    unsigned cum = 0, B = 0, above = 0;
    for (int b = NBINS - 1; b >= 0; --b) {
        unsigned nx = cum + hist[b];
        if (nx >= TK) { B = (unsigned)b; above = cum; break; }
        cum = nx;
        if (b == 0) { B = 0; above = cum; }
    }
    meta[0] = B;
    meta[1] = above;
    meta[4] = (TK > above) ? (TK - above) : 0u;
}

// ---------------------------------------------------------------------------
// K4: compact top-k candidates (unsorted) via atomic counters.
// ---------------------------------------------------------------------------
__global__ void compact_kernel(const float* __restrict__ scores, unsigned* __restrict__ meta,
                               float* __restrict__ tks, int* __restrict__ tki) {
    int a = blockIdx.x * blockDim.x + threadIdx.x;
    if (a >= NA) return;
    float s = scores[a];
    unsigned bin = orderKey(s) >> 20;
    unsigned B = meta[0], above = meta[1], need = meta[4];
    if (bin > B) {
        unsigned p = atomicAdd(&meta[2], 1u);
        if (p < TK) { tks[p] = s; tki[p] = a; }
    } else if (bin == B) {
        unsigned q = atomicAdd(&meta[3], 1u);
        if (q < need) { unsigned p = above + q; tks[p] = s; tki[p] = a; }
    }
}

// ---------------------------------------------------------------------------
// K5: single-workgroup bitonic sort of 8192 (score,idx) pairs, descending.
// 64KB LDS, 1024 threads (32 wave32 waves on one WGP).
// ---------------------------------------------------------------------------
__global__ __launch_bounds__(1024) void sort_topk_kernel(float* tks, int* tki) {
    __shared__ float ks[TKP];
    __shared__ int   vs[TKP];
    int t = threadIdx.x;
#pragma unroll
    for (int m = 0; m < 8; ++m) { int i = t + m * 1024; ks[i] = tks[i]; vs[i] = tki[i]; }
    for (int k = 2; k <= TKP; k <<= 1) {
        for (int j = k >> 1; j > 0; j >>= 1) {
            __syncthreads();
#pragma unroll
            for (int m = 0; m < 8; ++m) {
                int i = t + m * 1024;
                int ixj = i ^ j;
                if (ixj > i) {
                    bool up = ((i & k) == 0); // descending overall
                    float a = ks[i], b = ks[ixj];
                    if (up ? (a < b) : (a > b)) {
                        ks[i] = b; ks[ixj] = a;
                        int tv = vs[i]; vs[i] = vs[ixj]; vs[ixj] = tv;
                    }
                }
            }
        }
    }
    __syncthreads();
#pragma unroll
    for (int m = 0; m < 8; ++m) { int i = t + m * 1024; tks[i] = ks[i]; tki[i] = vs[i]; }
}

// ---------------------------------------------------------------------------
// K6: gather boxes (float4) + detections for sorted top-k.
// ---------------------------------------------------------------------------
__global__ void gather_topk_kernel(const float4* __restrict__ boxes, const float* __restrict__ det,
                                   const int* __restrict__ tki, float4* __restrict__ tkb,
                                   float* __restrict__ tkd) {
    int i = blockIdx.x * blockDim.x + threadIdx.x;
    if (i >= TKP) return;
    int src = tki[i];
    tkb[i] = boxes[src];
    if (i < TK) {
#pragma unroll
        for (int k = 0; k < DD; ++k) tkd[i * DD + k] = det[(size_t)src * DD + k];
    }
}

// ---------------------------------------------------------------------------
// K7: greedy NMS pass 1 over 8000 sorted boxes. Boxes staged global->LDS with
// async-to-LDS b128 copies (ASYNCcnt), suppression bitmask in LDS, early-exit
// at 300 accepted (provably equivalent to reference cumsum cap).
// ---------------------------------------------------------------------------
__global__ __launch_bounds__(1024) void nms1_kernel(const float4* __restrict__ tkb,
                                                    const float* __restrict__ tks,
                                                    const float* __restrict__ tkd,
                                                    float* __restrict__ s1, float4* __restrict__ b1,
                                                    float* __restrict__ d1) {
    __shared__ float4   lbox[TK];            // 128000 B (CDNA5: 320KB LDS per WGP)
    __shared__ unsigned keepw[(TK + 31) / 32]; // 250 words
    __shared__ int      sel[MB];
    __shared__ int      acc;
    int t = threadIdx.x;

#if defined(__AMDGCN__)
    // Async DMA of the 8000 boxes into LDS (tracked by ASYNCcnt).
#pragma unroll
    for (int m = 0; m < 8; ++m) {
        int i = t + m * 1024;
        if (i < TK) {
            unsigned lds_addr = (unsigned)(uintptr_t)(&lbox[i]);
            const float4* g = tkb + i;
            asm volatile("global_load_async_to_lds_b128 %0, %1, off"
                         :: "v"(lds_addr), "v"(g) : "memory");
        }
    }
    asm volatile("s_wait_asynccnt 0" ::: "memory");
#else
    for (int m = 0; m < 8; ++m) { int i = t + m * 1024; if (i < TK) lbox[i] = tkb[i]; }
#endif
    for (int w = t; w < (TK + 31) / 32; w += 1024) keepw[w] = 0u;
    if (t == 0) acc = 0;
    __syncthreads();
#pragma unroll
    for (int m = 0; m < 8; ++m) {
        int i = t + m * 1024;
        if (i < TK && tks[i] > SCORE_THR) atomicOr(&keepw[i >> 5], 1u << (i & 31));
    }
    __syncthreads();

    for (int i = 0; i < TK; ++i) {
        bool kept = (keepw[i >> 5] >> (i & 31)) & 1u; // uniform
        if (kept) {
            if (t == 0) sel[acc] = i;
            __syncthreads();
            float4 bi = lbox[i];
            float ai = boxArea(bi);
            for (int j = i + 1 + t; j < TK; j += 1024) {
                if ((keepw[j >> 5] >> (j & 31)) & 1u) {
                    float4 bj = lbox[j];
                    float ih = fmaxf(fminf(bi.z, bj.z) - fmaxf(bi.x, bj.x), 0.f);
                    float iw = fmaxf(fminf(bi.w, bj.w) - fmaxf(bi.y, bj.y), 0.f);
                    float inter = ih * iw;
                    float uni = fmaxf(ai + boxArea(bj) - inter, 1e-8f);
                    if (inter > NMS1_THR * uni) atomicAnd(&keepw[j >> 5], ~(1u << (j & 31)));
                }
            }
            if (t == 0) acc = acc + 1;
            __syncthreads();
            if (acc >= MB) break; // uniform
        }
    }
    __syncthreads();
    int cnt = acc;
    for (int r = t; r < MB; r += 1024) {
        bool valid = r < cnt;
        int src = valid ? sel[r] : 0;  // reference fills with rank 0
        s1[r] = valid ? tks[src] : NEGF;
        b1[r] = lbox[src];
#pragma unroll
        for (int k = 0; k < DD; ++k) d1[r * DD + k] = tkd[src * DD + k];
    }
}

// ---------------------------------------------------------------------------
// K8: 300x300 IoU matrix for pass 2, tiled 16x16, one wave per tile.
// The union's area_i + area_j outer-sum is computed with V_WMMA_F32_16X16X4_F32:
//   A(16x4) = [area_r | 1 | 0 | 0],  B(4x16) = [1 ; area_c ; 0 ; 0]  =>  D = area_r (+) area_c
// ---------------------------------------------------------------------------
__global__ __launch_bounds__(32) void iou2_tile_kernel(const float4* __restrict__ b1,
                                                       float* __restrict__ iou) {
    const int NT = (MB + 15) / 16; // 19
    int tile = blockIdx.x;
    int tr = tile / NT, tc = tile % NT;
    int lane = threadIdx.x;
    int r0 = tr * 16, c0 = tc * 16;
    int lr = r0 + (lane & 15), lc = c0 + (lane & 15);
    float4 brow = (lr < MB) ? b1[lr] : make_float4(0.f, 0.f, 0.f, 0.f);
    float4 bcol = (lc < MB) ? b1[lc] : make_float4(0.f, 0.f, 0.f, 0.f);
    float areaR = boxArea(brow), areaC = boxArea(bcol);
    bool lo = lane < 16;
    // A layout: lanes0-15 hold M, v0=K0, v1=K1; lanes16-31 -> K2,K3 (zero)
    v2f a; a.x = lo ? areaR : 0.f; a.y = lo ? 1.f : 0.f;
    // B layout mirrored: v0=K0 row (ones), v1=K1 row (area_c)
    v2f b; b.x = lo ? 1.f : 0.f;   b.y = lo ? areaC : 0.f;
    v8f c = {};
    v8f d = __builtin_amdgcn_wmma_f32_16x16x4_f32(
        /*neg_a=*/false, a, /*neg_b=*/false, b,
        /*c_mod=*/(short)0, c, /*reuse_a=*/false, /*reuse_b=*/false);
    // C/D layout: VGPR v, lane l -> M = v + 8*(l>=16), N = l&15
    int n = lane & 15;
    int mbase = (lane >= 16) ? 8 : 0;
    int gc = c0 + n;
#pragma unroll
    for (int v = 0; v < 8; ++v) {
        int gm = r0 + mbase + v;
        if (gm < MB && gc < MB) {
            float4 bi = b1[gm];
            float4 bj = b1[gc];
            float ih = fmaxf(fminf(bi.z, bj.z) - fmaxf(bi.x, bj.x), 0.f);
            float iw = fmaxf(fminf(bi.w, bj.w) - fmaxf(bi.y, bj.y), 0.f);
            float inter = ih * iw;
            float uni = fmaxf(d[v] - inter, 1e-8f); // d[v] = area_i + area_j (WMMA)
            iou[gm * MB + gc] = inter / uni;
        }
    }
}

// ---------------------------------------------------------------------------
// K9: greedy NMS pass 2 (n=300, iou_thr=0.05, score_thr=NEG) + final output:
// out[r] = d1[sel2[r]] for valid rows, zeros otherwise.
// ---------------------------------------------------------------------------
__global__ __launch_bounds__(1024) void nms2_kernel(const float* __restrict__ s1,
                                                    const float* __restrict__ iou,
                                                    const float* __restrict__ d1,
                                                    float* __restrict__ out) {
    __shared__ unsigned keepw[(MB + 31) / 32];
    __shared__ int sel[MB];
    __shared__ int acc;
    int t = threadIdx.x;
    if (t < (MB + 31) / 32) keepw[t] = 0u;
    if (t == 0) acc = 0;
    __syncthreads();
    if (t < MB && s1[t] > NEGF) atomicOr(&keepw[t >> 5], 1u << (t & 31));
    __syncthreads();
    for (int i = 0; i < MB; ++i) {
        bool kept = (keepw[i >> 5] >> (i & 31)) & 1u; // uniform
        if (kept) {
            if (t == 0) sel[acc] = i;
            __syncthreads();
            int j = i + 1 + t;
            if (j < MB && ((keepw[j >> 5] >> (j & 31)) & 1u) &&
                iou[i * MB + j] > NMS2_THR)
                atomicAnd(&keepw[j >> 5], ~(1u << (j & 31)));
            if (t == 0) acc = acc + 1;
            __syncthreads();
        }
    }
    __syncthreads();
    int cnt = acc;
    for (int e = t; e < MB * DD; e += 1024) {
        int r = e / DD, k = e % DD;
        out[e] = (r < cnt) ? d1[sel[r] * DD + k] : 0.f;
    }
}

// ---------------------------------------------------------------------------
extern "C" void kernel_launch(void* const* d_in, const int* in_sizes, int n_in,
                              void* d_out, int out_size, void* d_ws, size_t ws_size,
                              hipStream_t stream) {
    const float* boxes = (const float*)d_in[0];  // (1, NA, 4)
    const float* cls   = (const float*)d_in[1];  // (1, NA, 80)
    const float* det   = (const float*)d_in[2];  // (1, NA, 6)
    float* out = (float*)d_out;                  // (1, 300, 6) fp32
    char* ws = (char*)d_ws;

    float*    scores = (float*)(ws + OFF_SCORES);
    unsigned* hist   = (unsigned*)(ws + OFF_HIST);
    unsigned* meta   = (unsigned*)(ws + OFF_META);
    float*    tks    = (float*)(ws + OFF_TKS);
    int*      tki    = (int*)(ws + OFF_TKI);
    float4*   tkb    = (float4*)(ws + OFF_TKB);
    float*    tkd    = (float*)(ws + OFF_TKD);
    float*    s1     = (float*)(ws + OFF_S1);
    float4*   b1     = (float4*)(ws + OFF_B1);
    float*    d1     = (float*)(ws + OFF_D1);
    float*    iou2   = (float*)(ws + OFF_IOU2);

    init_ws_kernel<<<(TKP + 255) / 256, 256, 0, stream>>>(hist, meta, tks, tki);
    score_max_kernel<<<(NA * 4 + 255) / 256, 256, 0, stream>>>(cls, scores);
    hist_kernel<<<(NA + 255) / 256, 256, 0, stream>>>(scores, hist);
    threshold_kernel<<<1, 32, 0, stream>>>(hist, meta);
    compact_kernel<<<(NA + 255) / 256, 256, 0, stream>>>(scores, meta, tks, tki);
    sort_topk_kernel<<<1, 1024, 0, stream>>>(tks, tki);
    gather_topk_kernel<<<(TKP + 255) / 256, 256, 0, stream>>>((const float4*)boxes, det, tki, tkb, tkd);
    nms1_kernel<<<1, 1024, 0, stream>>>(tkb, tks, tkd, s1, b1, d1);
    iou2_tile_kernel<<<((MB + 15) / 16) * ((MB + 15) / 16), 32, 0, stream>>>(b1, iou2);
    nms2_kernel<<<1, 1024, 0, stream>>>(s1, iou2, d1, out);
}